// Attention_29867202576957
// MI455X (gfx1250) — compile-verified
//
#include <hip/hip_runtime.h>
#include <hip/hip_bf16.h>
#include <stdint.h>

// ---------------------------------------------------------------------------
// Attention block for MI455X (gfx1250), fp32 end-to-end via V_WMMA_F32_16X16X4_F32
//   DIM=1024, N_HEADS=16, N_KV_HEADS=4, HEAD_DIM=64, SEQ=4096, BSZ=1
// Workspace (floats): xq[4096x1024] | xk[4096x256] | xv[4096x256] | attn[4096x1024]
// ---------------------------------------------------------------------------

#define SEQ      4096
#define DIM      1024
#define NH       16
#define NKV      4
#define HD       64
#define HDH      32
#define KV_DIM   (NKV * HD)   // 256

typedef __attribute__((ext_vector_type(2))) float        v2f;
typedef __attribute__((ext_vector_type(8))) float        v8f;
typedef __attribute__((ext_vector_type(4))) unsigned int u32x4;
typedef __attribute__((ext_vector_type(4))) int          i32x4;
typedef __attribute__((ext_vector_type(8))) int          i32x8;

// D(16x16,f32) += A(16x4,f32) x B(4x16,f32)
#define WMMA_F32(a, b, c) \
  __builtin_amdgcn_wmma_f32_16x16x4_f32(false, (a), false, (b), (short)0, (c), false, false)

// ---------------------------------------------------------------------------
// Tensor Data Mover: async 2D tile load (rows x row_elems fp32, row stride in
// elements) from global to LDS. D# per CDNA5 ISA ch.8. Tracked by TENSORcnt.
// ---------------------------------------------------------------------------
__device__ __forceinline__ void tdm_load_tile_f32(uint64_t gaddr, uint32_t lds_off,
                                                  uint32_t rows, uint32_t row_elems,
                                                  uint32_t row_stride_elems) {
  u32x4 g0;
  g0.x = 1u;                                                // count=1, no gather
  g0.y = lds_off;                                           // lds_addr
  g0.z = (uint32_t)gaddr;                                   // global_addr[31:0]
  g0.w = ((uint32_t)(gaddr >> 32) & 0x1FFFFFFu) | (2u << 30); // addr[56:32] | type=2
  i32x8 g1;
  g1[0] = (int)(2u << 16);                                  // data_size = 4 bytes
  g1[1] = (int)(row_elems << 16);                           // tensor_dim0[15:0]
  g1[2] = (int)((row_elems >> 16) | (rows << 16));          // dim0[31:16] | dim1[15:0]
  g1[3] = (int)((rows >> 16) | (row_elems << 16));          // dim1[31:16] | tile_dim0
  g1[4] = (int)rows;                                        // tile_dim1 (tile_dim2=0)
  g1[5] = (int)row_stride_elems;                            // tensor_dim0_stride[31:0]
  g1[6] = 0;                                                // stride hi | dim1_stride lo
  g1[7] = 0;
  i32x4 gz = {0, 0, 0, 0};
#if __clang_major__ >= 23
  i32x8 gz8 = {0, 0, 0, 0, 0, 0, 0, 0};
  __builtin_amdgcn_tensor_load_to_lds(g0, g1, gz, gz, gz8, 0);
#else
  __builtin_amdgcn_tensor_load_to_lds(g0, g1, gz, gz, 0);
#endif
}

// ---------------------------------------------------------------------------
// GEMM:  C[M,N] = A[M,K] * W[N,K]^T      (row-major A, row-major W)
// One wave computes a 32x64 tile (2 M-subtiles x 4 N-subtiles, 8 accums),
// register double-buffered over K so loads for k+4 fly under step-k WMMAs.
// blockDim = (32,4); grid = (M/32, ceil(N/256))
// ---------------------------------------------------------------------------
__global__ __launch_bounds__(128)
void gemm_nt_wmma(const float* __restrict__ A, const float* __restrict__ W,
                  float* __restrict__ C, int M, int N, int K) {
  const int lane = threadIdx.x;
  const int wave = threadIdx.y;
  const int m0   = blockIdx.x * 32;
  const int n0   = (blockIdx.y * 4 + wave) * 64;
  if (n0 >= N) return;                    // wave-uniform

  const int half = lane >> 4;
  const int lid  = lane & 15;

  const float* A0 = A + (size_t)(m0 + lid) * K + 2 * half;
  const float* A1 = A0 + (size_t)16 * K;
  const float* W0 = W + (size_t)(n0 + lid) * K + 2 * half;
  const float* W1 = W0 + (size_t)16 * K;
  const float* W2 = W0 + (size_t)32 * K;
  const float* W3 = W0 + (size_t)48 * K;

  v8f acc[8] = {v8f{}, v8f{}, v8f{}, v8f{}, v8f{}, v8f{}, v8f{}, v8f{}};

  v2f a0 = *(const v2f*)(A0);
  v2f a1 = *(const v2f*)(A1);
  v2f b0 = *(const v2f*)(W0);
  v2f b1 = *(const v2f*)(W1);
  v2f b2 = *(const v2f*)(W2);
  v2f b3 = *(const v2f*)(W3);

  for (int k = 0; k + 4 < K; k += 4) {
    // issue next step's loads first: WMMAs below only need these later
    v2f na0 = *(const v2f*)(A0 + k + 4);
    v2f na1 = *(const v2f*)(A1 + k + 4);
    v2f nb0 = *(const v2f*)(W0 + k + 4);
    v2f nb1 = *(const v2f*)(W1 + k + 4);
    v2f nb2 = *(const v2f*)(W2 + k + 4);
    v2f nb3 = *(const v2f*)(W3 + k + 4);

    acc[0] = WMMA_F32(a0, b0, acc[0]);
    acc[1] = WMMA_F32(a0, b1, acc[1]);
    acc[2] = WMMA_F32(a0, b2, acc[2]);
    acc[3] = WMMA_F32(a0, b3, acc[3]);
    acc[4] = WMMA_F32(a1, b0, acc[4]);
    acc[5] = WMMA_F32(a1, b1, acc[5]);
    acc[6] = WMMA_F32(a1, b2, acc[6]);
    acc[7] = WMMA_F32(a1, b3, acc[7]);

    a0 = na0; a1 = na1; b0 = nb0; b1 = nb1; b2 = nb2; b3 = nb3;
  }
  acc[0] = WMMA_F32(a0, b0, acc[0]);
  acc[1] = WMMA_F32(a0, b1, acc[1]);
  acc[2] = WMMA_F32(a0, b2, acc[2]);
  acc[3] = WMMA_F32(a0, b3, acc[3]);
  acc[4] = WMMA_F32(a1, b0, acc[4]);
  acc[5] = WMMA_F32(a1, b1, acc[5]);
  acc[6] = WMMA_F32(a1, b2, acc[6]);
  acc[7] = WMMA_F32(a1, b3, acc[7]);

  // C/D layout: vgpr r, lane l -> row = r + 8*(l>=16), col = l%16
#pragma unroll
  for (int ms = 0; ms < 2; ++ms) {
#pragma unroll
    for (int t = 0; t < 4; ++t) {
#pragma unroll
      for (int r = 0; r < 8; ++r) {
        C[(size_t)(m0 + ms * 16 + r + 8 * half) * N + (n0 + t * 16 + lid)] =
            acc[ms * 4 + t][r];
      }
    }
  }
}

// ---------------------------------------------------------------------------
// RoPE, in place on x[s][h][d] with (cos,sin)[s][i], i < HD/2
// ---------------------------------------------------------------------------
__global__ void rope_kernel(float* __restrict__ x,
                            const float* __restrict__ cs,
                            const float* __restrict__ sn,
                            int nheads) {
  int idx = blockIdx.x * blockDim.x + threadIdx.x;
  int total = SEQ * nheads * HDH;
  if (idx >= total) return;
  int i = idx % HDH;
  int h = (idx / HDH) % nheads;
  int s = idx / (HDH * nheads);
  float c  = cs[s * HDH + i];
  float si = sn[s * HDH + i];
  float* p = x + ((size_t)s * nheads + h) * HD + 2 * i;
  float xr = p[0], xi = p[1];
  p[0] = xr * c - xi * si;
  p[1] = xr * si + xi * c;
}

// ---------------------------------------------------------------------------
// Flash attention (causal, GQA 4:1), one wave per (head, 16-row q tile).
// Q fragments hoisted into registers; V tiles DMA'd to LDS by the TDM and
// double-buffered (TENSORcnt), overlapping the next tile's transfer with the
// current tile's softmax + PV math.
// grid = (SEQ/16, NH), blockDim = 32.
// ---------------------------------------------------------------------------
__global__ __launch_bounds__(32)
void flash_attn_wmma(const float* __restrict__ xq, const float* __restrict__ xk,
                     const float* __restrict__ xv, float* __restrict__ out) {
  __shared__ float Plds[16 * 16];
  __shared__ float Vlds[2][16 * HD];

  const int qt  = blockIdx.x;
  const int h   = blockIdx.y;
  const int kvh = h / (NH / NKV);

  const int lane = threadIdx.x;
  const int half = lane >> 4;
  const int lid  = lane & 15;

  const float scale = 0.125f;           // 1/sqrt(64)

  // ---- hoist Q fragments (reused for every kv tile): 16 x v2f = 32 VGPRs ---
  const float* Qrow = xq + (size_t)(qt * 16 + lid) * DIM + h * HD + 2 * half;
  v2f qf[16];
#pragma unroll
  for (int j = 0; j < 16; ++j) qf[j] = *(const v2f*)(Qrow + 4 * j);

  const uint64_t vbase = (uint64_t)(uintptr_t)(xv + (size_t)kvh * HD);
  const uint32_t vlds0 = (uint32_t)(uintptr_t)(&Vlds[0][0]);
  const uint32_t vlds1 = (uint32_t)(uintptr_t)(&Vlds[1][0]);

  // kick off TDM for V tile 0
  tdm_load_tile_f32(vbase, vlds0, 16, HD, KV_DIM);

  float m_i[8], l_i[8];
  v8f   oacc[4] = {v8f{}, v8f{}, v8f{}, v8f{}};
#pragma unroll
  for (int r = 0; r < 8; ++r) { m_i[r] = -1e30f; l_i[r] = 0.0f; }

  for (int kt = 0; kt <= qt; ++kt) {
    // prefetch next V tile while we compute on this one
    if (kt < qt) {
      uint64_t g = vbase + (uint64_t)(kt + 1) * 16 * KV_DIM * sizeof(float);
      tdm_load_tile_f32(g, ((kt + 1) & 1) ? vlds1 : vlds0, 16, HD, KV_DIM);
    }

    // ---- S = Q x K^T (16x16) ------------------------------------------
    const float* Krow = xk + (size_t)(kt * 16 + lid) * KV_DIM + kvh * HD + 2 * half;
    v8f s = v8f{};
#pragma unroll
    for (int j = 0; j < 16; ++j) {
      v2f b = *(const v2f*)(Krow + 4 * j);
      s = WMMA_F32(qf[j], b, s);
    }

    // ---- scale + causal mask ------------------------------------------
    float sv[8];
#pragma unroll
    for (int r = 0; r < 8; ++r) {
      int qg = qt * 16 + r + 8 * half;
      int kg = kt * 16 + lid;
      float v = s[r] * scale;
      sv[r] = (kg > qg) ? -1e30f : v;
    }

    // ---- online softmax (row reductions across each 16-lane half) -----
    float fac[8], p[8];
#pragma unroll
    for (int r = 0; r < 8; ++r) {
      float v = sv[r];
      v = fmaxf(v, __shfl_xor(v, 1, 32));
      v = fmaxf(v, __shfl_xor(v, 2, 32));
      v = fmaxf(v, __shfl_xor(v, 4, 32));
      v = fmaxf(v, __shfl_xor(v, 8, 32));
      float mnew = fmaxf(m_i[r], v);
      fac[r] = __expf(m_i[r] - mnew);
      m_i[r] = mnew;
      p[r]   = __expf(sv[r] - mnew);
    }
#pragma unroll
    for (int r = 0; r < 8; ++r) {
      float v = p[r];
      v += __shfl_xor(v, 1, 32);
      v += __shfl_xor(v, 2, 32);
      v += __shfl_xor(v, 4, 32);
      v += __shfl_xor(v, 8, 32);
      l_i[r] = l_i[r] * fac[r] + v;
    }
#pragma unroll
    for (int t = 0; t < 4; ++t)
#pragma unroll
      for (int r = 0; r < 8; ++r) oacc[t][r] *= fac[r];

    // ---- stage P through LDS: C-layout -> A-layout --------------------
#pragma unroll
    for (int r = 0; r < 8; ++r) Plds[(r + 8 * half) * 16 + lid] = p[r];
    __syncthreads();

    // ---- wait for this kv tile's V DMA (older of <=2 outstanding) -----
    // (immediate must be a literal constant; branch is wave-uniform)
    if (kt < qt) __builtin_amdgcn_s_wait_tensorcnt(1);
    else         __builtin_amdgcn_s_wait_tensorcnt(0);
    const float* Vt = &Vlds[kt & 1][0];

    // ---- O += P x V ----------------------------------------------------
#pragma unroll
    for (int t = 0; t < 4; ++t) {
#pragma unroll
      for (int k = 0; k < 16; k += 4) {
        v2f a = *(const v2f*)(&Plds[lid * 16 + k + 2 * half]);
        v2f b;
        b.x = Vt[(k + 2 * half + 0) * HD + t * 16 + lid];
        b.y = Vt[(k + 2 * half + 1) * HD + t * 16 + lid];
        oacc[t] = WMMA_F32(a, b, oacc[t]);
      }
    }
    __syncthreads();   // P / V buffers reused next iteration
  }

  // ---- normalize and write attn output [SEQ][DIM] ----------------------
#pragma unroll
  for (int t = 0; t < 4; ++t) {
#pragma unroll
    for (int r = 0; r < 8; ++r) {
      int qg = qt * 16 + r + 8 * half;
      out[(size_t)qg * DIM + h * HD + t * 16 + lid] = oacc[t][r] / l_i[r];
    }
  }
}

// ---------------------------------------------------------------------------
extern "C" void kernel_launch(void* const* d_in, const int* in_sizes, int n_in,
                              void* d_out, int out_size, void* d_ws, size_t ws_size,
                              hipStream_t stream) {
  const float* x  = (const float*)d_in[0];
  const float* fc = (const float*)d_in[1];
  const float* fs = (const float*)d_in[2];
  const float* wq = (const float*)d_in[3];
  const float* wk = (const float*)d_in[4];
  const float* wv = (const float*)d_in[5];
  const float* wo = (const float*)d_in[6];

  float* xq   = (float*)d_ws;                       // 4096 x 1024
  float* xk   = xq + (size_t)SEQ * DIM;             // 4096 x 256
  float* xv   = xk + (size_t)SEQ * KV_DIM;          // 4096 x 256
  float* attn = xv + (size_t)SEQ * KV_DIM;          // 4096 x 1024
  float* outp = (float*)d_out;

  dim3 gblk(32, 4);

  gemm_nt_wmma<<<dim3(SEQ / 32, DIM / 256), gblk, 0, stream>>>(x, wq, xq, SEQ, DIM,    DIM);
  gemm_nt_wmma<<<dim3(SEQ / 32, 1),         gblk, 0, stream>>>(x, wk, xk, SEQ, KV_DIM, DIM);
  gemm_nt_wmma<<<dim3(SEQ / 32, 1),         gblk, 0, stream>>>(x, wv, xv, SEQ, KV_DIM, DIM);

  {
    int tq = SEQ * NH * HDH, tk = SEQ * NKV * HDH;
    rope_kernel<<<(tq + 255) / 256, 256, 0, stream>>>(xq, fc, fs, NH);
    rope_kernel<<<(tk + 255) / 256, 256, 0, stream>>>(xk, fc, fs, NKV);
  }

  flash_attn_wmma<<<dim3(SEQ / 16, NH), 32, 0, stream>>>(xq, xk, xv, attn);

  gemm_nt_wmma<<<dim3(SEQ / 32, DIM / 256), gblk, 0, stream>>>(attn, wo, outp, SEQ, DIM, DIM);
}